// Class_net_55533927137974
// MI455X (gfx1250) — compile-verified
//
#include <hip/hip_runtime.h>
#include <hip/hip_bf16.h>
#include <math.h>

#define N_NODES 100000
#define N_EDGES 3200000
#define E_TOT   (N_EDGES + N_NODES)
#define NEG_SLOPE 0.2f

typedef float v2f __attribute__((ext_vector_type(2)));
typedef float v8f __attribute__((ext_vector_type(8)));

// float atomic max via signed-max / unsigned-min trick (valid for mixed signs,
// init to -inf). Lowers to global_atomic_max_i32 / global_atomic_min_u32.
__device__ __forceinline__ void atomicMaxF(float* addr, float val) {
    if (val >= 0.0f) atomicMax((int*)addr, __float_as_int(val));
    else             atomicMin((unsigned int*)addr, __float_as_uint(val));
}

__global__ void fill_kernel(float* __restrict__ p, float v, int n) {
    int i = blockIdx.x * blockDim.x + threadIdx.x;
    if (i < n) p[i] = v;
}

// ---------------------------------------------------------------------------
// H[N,64] = X[N,512] @ W[512,64] using V_WMMA_F32_16X16X4_F32 (fp32 matrix pipe).
// One wave computes a full 16x64 output stripe: the A fragment (16x4 of x) is
// loaded ONCE per k-step and reused across 4 WMMAs (4 N-tiles, 4 accumulators),
// quartering x traffic vs. one-tile-per-wave.
// A frag (16x4 f32): lanes 0-15 -> M rows, v0/v1 = K, K+1; lanes 16-31 -> K+2, K+3.
// B frag mirrors with N columns. C: 8 VGPRs, lane half selects M / M+8.
// ---------------------------------------------------------------------------
__global__ void __launch_bounds__(256)
gemm512x64_wmma(const float* __restrict__ X,
                const float* __restrict__ W,
                float* __restrict__ H) {
    const int K = 512, NC = 64;
    int wave = (blockIdx.x * blockDim.x + threadIdx.x) >> 5;
    int lane = threadIdx.x & 31;
    int m0 = wave * 16;
    if (m0 >= N_NODES) return;
    int r = lane & 15, half = lane >> 4;
    const float* xrow = X + (size_t)(m0 + r) * K + half * 2;
    const float* wrow = W + r + (size_t)(half * 2) * NC;   // W[(k+2*half)][r + n0]
    v8f c0 = {}, c1 = {}, c2 = {}, c3 = {};
#pragma unroll 2
    for (int k = 0; k < K; k += 4) {
        v2f a;
        a.x = xrow[k];
        a.y = xrow[k + 1];
        const float* wk = wrow + (size_t)k * NC;
        v2f b0, b1, b2, b3;
        b0.x = wk[0];       b0.y = wk[NC + 0];
        b1.x = wk[16];      b1.y = wk[NC + 16];
        b2.x = wk[32];      b2.y = wk[NC + 32];
        b3.x = wk[48];      b3.y = wk[NC + 48];
        c0 = __builtin_amdgcn_wmma_f32_16x16x4_f32(false, a, false, b0, (short)0, c0, false, false);
        c1 = __builtin_amdgcn_wmma_f32_16x16x4_f32(false, a, false, b1, (short)0, c1, false, false);
        c2 = __builtin_amdgcn_wmma_f32_16x16x4_f32(false, a, false, b2, (short)0, c2, false, false);
        c3 = __builtin_amdgcn_wmma_f32_16x16x4_f32(false, a, false, b3, (short)0, c3, false, false);
    }
#pragma unroll
    for (int v = 0; v < 8; ++v) {
        float* hr = H + (size_t)(m0 + half * 8 + v) * NC + r;
        hr[0]  = c0[v];
        hr[16] = c1[v];
        hr[32] = c2[v];
        hr[48] = c3[v];
    }
}

// Z[N,16] = Hin[N,64] @ W[64,16], single N-tile per wave.
__global__ void __launch_bounds__(256)
gemm64x16_wmma(const float* __restrict__ Hin,
               const float* __restrict__ W,
               float* __restrict__ Z) {
    const int K = 64, NC = 16;
    int wave = (blockIdx.x * blockDim.x + threadIdx.x) >> 5;
    int lane = threadIdx.x & 31;
    int m0 = wave * 16;
    if (m0 >= N_NODES) return;
    int r = lane & 15, half = lane >> 4;
    const float* xrow = Hin + (size_t)(m0 + r) * K + half * 2;
    const float* wcol = W + r + (size_t)(half * 2) * NC;
    v8f c = {};
#pragma unroll
    for (int k = 0; k < K; k += 4) {
        v2f a, b;
        a.x = xrow[k];
        a.y = xrow[k + 1];
        b.x = wcol[(size_t)k * NC];
        b.y = wcol[(size_t)(k + 1) * NC];
        c = __builtin_amdgcn_wmma_f32_16x16x4_f32(false, a, false, b,
                                                  (short)0, c, false, false);
    }
#pragma unroll
    for (int v = 0; v < 8; ++v)
        Z[(size_t)(m0 + half * 8 + v) * NC + r] = c[v];
}

// a_src[n,h] = dot(h[n,h,:], att_src[h,:]) ; same for a_dst. 8 heads x 8 ch.
__global__ void att_dots1(const float* __restrict__ H,
                          const float* __restrict__ attS,
                          const float* __restrict__ attD,
                          float* __restrict__ aS, float* __restrict__ aD) {
    int t = blockIdx.x * blockDim.x + threadIdx.x;
    if (t >= N_NODES * 8) return;
    int n = t >> 3, hh = t & 7;
    const float* hp = H + (size_t)n * 64 + hh * 8;
    float s = 0.f, d = 0.f;
#pragma unroll
    for (int c = 0; c < 8; ++c) {
        s += hp[c] * attS[hh * 8 + c];
        d += hp[c] * attD[hh * 8 + c];
    }
    aS[t] = s; aD[t] = d;
}

__device__ __forceinline__ void edge_sd(int e, const int* src, const int* dst,
                                        int& s, int& d) {
    if (e < N_EDGES) { s = src[e]; d = dst[e]; }
    else             { s = d = e - N_EDGES; }   // self loops
}

// ---- layer 1 edge passes: (edge, head) per thread --------------------------
__global__ void edge1_max(const int* __restrict__ src, const int* __restrict__ dst,
                          const float* __restrict__ aS, const float* __restrict__ aD,
                          float* __restrict__ m) {
    int t = blockIdx.x * blockDim.x + threadIdx.x;
    if (t >= E_TOT * 8) return;
    int e = t >> 3, hh = t & 7, s, d;
    edge_sd(e, src, dst, s, d);
    float v = aS[s * 8 + hh] + aD[d * 8 + hh];
    v = v > 0.f ? v : NEG_SLOPE * v;
    atomicMaxF(&m[d * 8 + hh], v);
}

__global__ void edge1_sum(const int* __restrict__ src, const int* __restrict__ dst,
                          const float* __restrict__ aS, const float* __restrict__ aD,
                          const float* __restrict__ m, float* __restrict__ ssum) {
    int t = blockIdx.x * blockDim.x + threadIdx.x;
    if (t >= E_TOT * 8) return;
    int e = t >> 3, hh = t & 7, s, d;
    edge_sd(e, src, dst, s, d);
    float v = aS[s * 8 + hh] + aD[d * 8 + hh];
    v = v > 0.f ? v : NEG_SLOPE * v;
    atomicAdd(&ssum[d * 8 + hh], __expf(v - m[d * 8 + hh]));
}

__global__ void edge1_agg(const int* __restrict__ src, const int* __restrict__ dst,
                          const float* __restrict__ aS, const float* __restrict__ aD,
                          const float* __restrict__ m, const float* __restrict__ ssum,
                          const float* __restrict__ H, float* __restrict__ out) {
    int t = blockIdx.x * blockDim.x + threadIdx.x;
    if (t >= E_TOT * 8) return;
    int e = t >> 3, hh = t & 7, s, d;
    edge_sd(e, src, dst, s, d);
    const float* hp = H + (size_t)s * 64 + hh * 8;
    __builtin_prefetch(hp, 0, 0);                    // global_prefetch_b8
    float v = aS[s * 8 + hh] + aD[d * 8 + hh];
    v = v > 0.f ? v : NEG_SLOPE * v;
    float alpha = __expf(v - m[d * 8 + hh]) / (ssum[d * 8 + hh] + 1e-16f);
    float* op = out + (size_t)d * 64 + hh * 8;
#pragma unroll
    for (int c = 0; c < 8; ++c) atomicAdd(&op[c], hp[c] * alpha);
}

// out1 += b1, then ELU, in place. Result feeds layer 2.
__global__ void bias_elu(float* __restrict__ o, const float* __restrict__ b) {
    int i = blockIdx.x * blockDim.x + threadIdx.x;
    if (i >= N_NODES * 64) return;
    float v = o[i] + b[i & 63];
    o[i] = v > 0.f ? v : (__expf(v) - 1.f);
}

// layer-2 per-node attention dots (1 head, 16 ch)
__global__ void att_dots2(const float* __restrict__ Z,
                          const float* __restrict__ attS,
                          const float* __restrict__ attD,
                          float* __restrict__ aS, float* __restrict__ aD) {
    int n = blockIdx.x * blockDim.x + threadIdx.x;
    if (n >= N_NODES) return;
    const float* zp = Z + (size_t)n * 16;
    float s = 0.f, d = 0.f;
#pragma unroll
    for (int c = 0; c < 16; ++c) { s += zp[c] * attS[c]; d += zp[c] * attD[c]; }
    aS[n] = s; aD[n] = d;
}

__global__ void edge2_max(const int* __restrict__ src, const int* __restrict__ dst,
                          const float* __restrict__ aS, const float* __restrict__ aD,
                          float* __restrict__ m) {
    int e = blockIdx.x * blockDim.x + threadIdx.x;
    if (e >= E_TOT) return;
    int s, d; edge_sd(e, src, dst, s, d);
    float v = aS[s] + aD[d];
    v = v > 0.f ? v : NEG_SLOPE * v;
    atomicMaxF(&m[d], v);
}

__global__ void edge2_sum(const int* __restrict__ src, const int* __restrict__ dst,
                          const float* __restrict__ aS, const float* __restrict__ aD,
                          const float* __restrict__ m, float* __restrict__ ssum) {
    int e = blockIdx.x * blockDim.x + threadIdx.x;
    if (e >= E_TOT) return;
    int s, d; edge_sd(e, src, dst, s, d);
    float v = aS[s] + aD[d];
    v = v > 0.f ? v : NEG_SLOPE * v;
    atomicAdd(&ssum[d], __expf(v - m[d]));
}

__global__ void edge2_agg(const int* __restrict__ src, const int* __restrict__ dst,
                          const float* __restrict__ aS, const float* __restrict__ aD,
                          const float* __restrict__ m, const float* __restrict__ ssum,
                          const float* __restrict__ Z, float* __restrict__ out) {
    int e = blockIdx.x * blockDim.x + threadIdx.x;
    if (e >= E_TOT) return;
    int s, d; edge_sd(e, src, dst, s, d);
    float v = aS[s] + aD[d];
    v = v > 0.f ? v : NEG_SLOPE * v;
    float alpha = __expf(v - m[d]) / (ssum[d] + 1e-16f);
    const float* zp = Z + (size_t)s * 16;
    float* op = out + (size_t)d * 16;
#pragma unroll
    for (int c = 0; c < 16; ++c) atomicAdd(&op[c], zp[c] * alpha);
}

__global__ void final_logsoftmax(const float* __restrict__ o,
                                 const float* __restrict__ b,
                                 float* __restrict__ y) {
    int n = blockIdx.x * blockDim.x + threadIdx.x;
    if (n >= N_NODES) return;
    float z[16], mx = -INFINITY;
#pragma unroll
    for (int c = 0; c < 16; ++c) { z[c] = o[n * 16 + c] + b[c]; mx = fmaxf(mx, z[c]); }
    float s = 0.f;
#pragma unroll
    for (int c = 0; c < 16; ++c) s += __expf(z[c] - mx);
    float l = __logf(s);
#pragma unroll
    for (int c = 0; c < 16; ++c) y[n * 16 + c] = z[c] - mx - l;
}

extern "C" void kernel_launch(void* const* d_in, const int* in_sizes, int n_in,
                              void* d_out, int out_size, void* d_ws, size_t ws_size,
                              hipStream_t stream) {
    const float* x     = (const float*)d_in[0];
    const int*   ei    = (const int*)  d_in[1];   // [2, E]: row0 = src, row1 = dst
    const float* W1    = (const float*)d_in[2];
    const float* attS1 = (const float*)d_in[3];
    const float* attD1 = (const float*)d_in[4];
    const float* b1    = (const float*)d_in[5];
    const float* W2    = (const float*)d_in[6];
    const float* attS2 = (const float*)d_in[7];
    const float* attD2 = (const float*)d_in[8];
    const float* b2    = (const float*)d_in[9];
    const int* src = ei;
    const int* dst = ei + N_EDGES;

    // workspace layout (floats): ~78.4 MB total
    float* ws   = (float*)d_ws;
    float* bufH = ws;                           // N*64  layer-1 features h
    float* aS1  = bufH + (size_t)N_NODES * 64;  // N*8
    float* aD1  = aS1 + (size_t)N_NODES * 8;    // N*8
    float* m1   = aD1 + (size_t)N_NODES * 8;    // N*8
    float* s1   = m1  + (size_t)N_NODES * 8;    // N*8
    float* out1 = s1  + (size_t)N_NODES * 8;    // N*64
    float* z2   = out1 + (size_t)N_NODES * 64;  // N*16
    float* aS2  = z2  + (size_t)N_NODES * 16;   // N
    float* aD2  = aS2 + N_NODES;                // N
    float* m2   = aD2 + N_NODES;                // N
    float* s2   = m2  + N_NODES;                // N  (out2 follows, contiguous)
    float* out2 = s2  + N_NODES;                // N*16
    float* y    = (float*)d_out;

    const int BT = 256;
    auto blk = [](long n, int t) { return (int)((n + t - 1) / t); };

    // init: m arrays -> -inf, s+out regions (contiguous) -> 0
    fill_kernel<<<blk(N_NODES * 8, BT), BT, 0, stream>>>(m1, -INFINITY, N_NODES * 8);
    fill_kernel<<<blk((long)N_NODES * 72, BT), BT, 0, stream>>>(s1, 0.f, N_NODES * 72);
    fill_kernel<<<blk(N_NODES, BT), BT, 0, stream>>>(m2, -INFINITY, N_NODES);
    fill_kernel<<<blk((long)N_NODES * 17, BT), BT, 0, stream>>>(s2, 0.f, N_NODES * 17);

    // layer 1: ceil(N/16) waves, each producing a 16x64 stripe
    {
        long threads = (long)((N_NODES + 15) / 16) * 32;
        gemm512x64_wmma<<<blk(threads, BT), BT, 0, stream>>>(x, W1, bufH);
    }
    att_dots1<<<blk((long)N_NODES * 8, BT), BT, 0, stream>>>(bufH, attS1, attD1, aS1, aD1);
    {
        int g = blk((long)E_TOT * 8, BT);
        edge1_max<<<g, BT, 0, stream>>>(src, dst, aS1, aD1, m1);
        edge1_sum<<<g, BT, 0, stream>>>(src, dst, aS1, aD1, m1, s1);
        edge1_agg<<<g, BT, 0, stream>>>(src, dst, aS1, aD1, m1, s1, bufH, out1);
    }
    bias_elu<<<blk((long)N_NODES * 64, BT), BT, 0, stream>>>(out1, b1);

    // layer 2
    {
        long threads = (long)((N_NODES + 15) / 16) * 32;
        gemm64x16_wmma<<<blk(threads, BT), BT, 0, stream>>>(out1, W2, z2);
    }
    att_dots2<<<blk(N_NODES, BT), BT, 0, stream>>>(z2, attS2, attD2, aS2, aD2);
    {
        int g = blk(E_TOT, BT);
        edge2_max<<<g, BT, 0, stream>>>(src, dst, aS2, aD2, m2);
        edge2_sum<<<g, BT, 0, stream>>>(src, dst, aS2, aD2, m2, s2);
        edge2_agg<<<g, BT, 0, stream>>>(src, dst, aS2, aD2, m2, s2, z2, out2);
    }
    final_logsoftmax<<<blk(N_NODES, BT), BT, 0, stream>>>(out2, b2, y);
}